// SelfAttention_53343493816717
// MI455X (gfx1250) — compile-verified
//
#include <hip/hip_runtime.h>
#include <hip/hip_bf16.h>
#include <math.h>

typedef float v2f __attribute__((ext_vector_type(2)));
typedef float v8f __attribute__((ext_vector_type(8)));

#define NEG_MIN_F (-65504.0f)
#define INV_SQRT_D 0.08838834764831845f   // 1/sqrt(128)

// ---------------------------------------------------------------------------
// C(8 x N) = X(8 x K) @ W(K x N) + bias, K = 4096, via V_WMMA_F32_16X16X4_F32.
// One 16-wide N tile per block; 8 waves split K (512 each = 128 WMMA steps);
// partial 16x16 tiles reduced through LDS.
// A layout (16x4 f32): lanes 0-15 K={kk,kk+1}, lanes 16-31 K={kk+2,kk+3}.
// B layout (4x16 f32): mirrored; column n0+lr per lane.
// ---------------------------------------------------------------------------
__global__ __launch_bounds__(256) void gemm8_wmma(
    const float* __restrict__ X, const float* __restrict__ W,
    const float* __restrict__ bias, float* __restrict__ out, int N)
{
    const int K = 4096;
    const int n0   = blockIdx.x * 16;
    const int w    = threadIdx.x >> 5;
    const int lane = threadIdx.x & 31;
    const int half = lane >> 4;
    const int lr   = lane & 15;
    const int row  = lr & 7;
    const float amask = (lr < 8) ? 1.0f : 0.0f;   // zero padded rows 8..15

    v8f c = {};
    const int kbeg = w * (K / 8);
    const float* xrow = X + (size_t)row * K;

    #pragma unroll 4
    for (int kk = kbeg; kk < kbeg + (K / 8); kk += 4) {
        const int ka = kk + 2 * half;
        v2f a, b;
        const float2 xv = *(const float2*)(xrow + ka);
        a.x = xv.x * amask;
        a.y = xv.y * amask;
        b.x = W[(size_t)ka * N + n0 + lr];
        b.y = W[(size_t)(ka + 1) * N + n0 + lr];
        c = __builtin_amdgcn_wmma_f32_16x16x4_f32(
                false, a, false, b, (short)0, c, false, false);
    }

    __shared__ float red[8][8][16];
    if (half == 0) {
        #pragma unroll
        for (int m = 0; m < 8; ++m) red[w][m][lr] = c[m];
    }
    __syncthreads();

    if (threadIdx.x < 128) {
        const int m = threadIdx.x >> 4;
        const int n = threadIdx.x & 15;
        float s = 0.0f;
        #pragma unroll
        for (int w2 = 0; w2 < 8; ++w2) s += red[w2][m][n];
        out[(size_t)m * N + n0 + n] = s + bias[n0 + n];
    }
}

// ---------------------------------------------------------------------------
// RoPE q in place; RoPE k -> cache row 4096; copy v -> cache row 4096.
// grid = 8 (batch), block = 256.
// ---------------------------------------------------------------------------
__global__ __launch_bounds__(256) void rope_append(
    float* __restrict__ qbuf, const float* __restrict__ kbuf,
    const float* __restrict__ vbuf, const int* __restrict__ pos,
    const float* __restrict__ cosT, const float* __restrict__ sinT,
    float* __restrict__ kout, float* __restrict__ vout)
{
    const int b = blockIdx.x;
    const int p = pos[b];
    const float* crow = cosT + (size_t)p * 64;
    const float* srow = sinT + (size_t)p * 64;

    // q: 32 heads x 32 rotary pairs
    for (int u = threadIdx.x; u < 1024; u += 256) {
        const int h = u >> 5, j = u & 31;
        float* qp = qbuf + (size_t)b * 4096 + (size_t)h * 128;
        const float q0 = qp[j], q1 = qp[j + 32];
        qp[j]      = q0 * crow[j]      - q1 * srow[j];
        qp[j + 32] = q1 * crow[j + 32] + q0 * srow[j + 32];
    }
    // k: 8 kv heads x 32 rotary pairs -> d_out row 4096
    {
        const int kvh = threadIdx.x >> 5, j = threadIdx.x & 31;
        const float* kp = kbuf + (size_t)b * 1024 + (size_t)kvh * 128;
        float* ko = kout + ((size_t)(b * 8 + kvh) * 4097 + 4096) * 128;
        const float k0 = kp[j], k1 = kp[j + 32];
        ko[j]      = k0 * crow[j]      - k1 * srow[j];
        ko[j + 32] = k1 * crow[j + 32] + k0 * srow[j + 32];
    }
    // k pass-through dims 64..127
    for (int u = threadIdx.x; u < 512; u += 256) {
        const int kvh = u >> 6, d = 64 + (u & 63);
        kout[((size_t)(b * 8 + kvh) * 4097 + 4096) * 128 + d] =
            kbuf[(size_t)b * 1024 + (size_t)kvh * 128 + d];
    }
    // v copy
    for (int u = threadIdx.x; u < 1024; u += 256) {
        const int kvh = u >> 7, d = u & 127;
        vout[((size_t)(b * 8 + kvh) * 4097 + 4096) * 128 + d] =
            vbuf[(size_t)b * 1024 + (size_t)kvh * 128 + d];
    }
}

// ---------------------------------------------------------------------------
// Fused cache-copy + flash-decode partials.
// grid = 64 (b,kv) x 8 K-splits = 512 blocks; block = 256 (8 waves).
// Each wave streams rows j (512B K + 512B V, float4/lane, coalesced),
// writes them to d_out (the concat copy), and online-softmaxes 4 GQA heads.
// Partials (m, l, acc[128]) spilled per (b,kv,head,split) to ws.
// ---------------------------------------------------------------------------
__global__ __launch_bounds__(256) void attn_partial(
    const float* __restrict__ kc, const float* __restrict__ vc,
    float* __restrict__ ko, float* __restrict__ vo,
    const float* __restrict__ qbuf, const float* __restrict__ maskp,
    float* __restrict__ part)
{
    const int blk   = blockIdx.x;
    const int bk    = blk >> 3;          // (b*8 + kv)
    const int split = blk & 7;
    const int b     = bk >> 3;
    const int kvh   = bk & 7;
    const int w     = threadIdx.x >> 5;
    const int lane  = threadIdx.x & 31;

    float4 qv[4];
    #pragma unroll
    for (int hh = 0; hh < 4; ++hh) {
        qv[hh] = *(const float4*)(qbuf + (size_t)b * 4096 +
                                  (size_t)(kvh * 4 + hh) * 128 + 4 * lane);
        qv[hh].x *= INV_SQRT_D; qv[hh].y *= INV_SQRT_D;
        qv[hh].z *= INV_SQRT_D; qv[hh].w *= INV_SQRT_D;
    }

    float m[4], l[4], acc[4][4];
    #pragma unroll
    for (int hh = 0; hh < 4; ++hh) {
        m[hh] = -INFINITY; l[hh] = 0.0f;
        acc[hh][0] = acc[hh][1] = acc[hh][2] = acc[hh][3] = 0.0f;
    }

    const size_t base  = (size_t)bk * 4096 * 128;
    const size_t obase = (size_t)bk * 4097 * 128;
    const int j0 = split * 512;

    for (int j = j0 + w; j < j0 + 512; j += 8) {
        const size_t off  = base  + (size_t)j * 128 + 4 * lane;
        const size_t ooff = obase + (size_t)j * 128 + 4 * lane;
        const float4 k4 = *(const float4*)(kc + off);
        const float4 v4 = *(const float4*)(vc + off);
        *(float4*)(ko + ooff) = k4;      // fused cache concat
        *(float4*)(vo + ooff) = v4;
        const float mv = maskp[(size_t)b * 4097 + j];
        const float madd = (mv == 1.0f) ? 0.0f : NEG_MIN_F;

        #pragma unroll
        for (int hh = 0; hh < 4; ++hh) {
            float s = qv[hh].x * k4.x + qv[hh].y * k4.y +
                      qv[hh].z * k4.z + qv[hh].w * k4.w;
            s += __shfl_xor(s, 16); s += __shfl_xor(s, 8);
            s += __shfl_xor(s, 4);  s += __shfl_xor(s, 2);
            s += __shfl_xor(s, 1);
            s += madd;
            const float mn   = fmaxf(m[hh], s);
            const float corr = __expf(m[hh] - mn);
            const float p    = __expf(s - mn);
            l[hh] = l[hh] * corr + p;
            acc[hh][0] = acc[hh][0] * corr + p * v4.x;
            acc[hh][1] = acc[hh][1] * corr + p * v4.y;
            acc[hh][2] = acc[hh][2] * corr + p * v4.z;
            acc[hh][3] = acc[hh][3] * corr + p * v4.w;
            m[hh] = mn;
        }
    }

    __shared__ float red[8][4][132];
    #pragma unroll
    for (int hh = 0; hh < 4; ++hh) {
        red[w][hh][4 * lane + 0] = acc[hh][0];
        red[w][hh][4 * lane + 1] = acc[hh][1];
        red[w][hh][4 * lane + 2] = acc[hh][2];
        red[w][hh][4 * lane + 3] = acc[hh][3];
        if (lane == 0) { red[w][hh][128] = m[hh]; red[w][hh][129] = l[hh]; }
    }
    __syncthreads();

    if (w < 4) {
        const int hh = w;
        float M = -INFINITY;
        #pragma unroll
        for (int w2 = 0; w2 < 8; ++w2) M = fmaxf(M, red[w2][hh][128]);
        float L = 0.0f, A0 = 0, A1 = 0, A2 = 0, A3 = 0;
        #pragma unroll
        for (int w2 = 0; w2 < 8; ++w2) {
            const float cc = __expf(red[w2][hh][128] - M);
            L  += red[w2][hh][129] * cc;
            A0 += red[w2][hh][4 * lane + 0] * cc;
            A1 += red[w2][hh][4 * lane + 1] * cc;
            A2 += red[w2][hh][4 * lane + 2] * cc;
            A3 += red[w2][hh][4 * lane + 3] * cc;
        }
        float* pp = part + (((size_t)bk * 4 + hh) * 8 + split) * 132;
        pp[4 * lane + 0] = A0; pp[4 * lane + 1] = A1;
        pp[4 * lane + 2] = A2; pp[4 * lane + 3] = A3;
        if (lane == 0) { pp[128] = M; pp[129] = L; }
    }
}

// ---------------------------------------------------------------------------
// Merge 8 split partials per (b,h), fold in appended key (row 4096), write
// attn rows laid out (B, H*D) for the output projection.
// grid = 64 (b,kv); block = 128 (4 waves = 4 heads).
// ---------------------------------------------------------------------------
__global__ __launch_bounds__(128) void attn_reduce(
    const float* __restrict__ part, const float* __restrict__ qbuf,
    const float* __restrict__ maskp, const float* __restrict__ ko,
    const float* __restrict__ vo, float* __restrict__ attnbuf)
{
    const int bk = blockIdx.x;
    const int b = bk >> 3, kvh = bk & 7;
    const int w = threadIdx.x >> 5, lane = threadIdx.x & 31;
    const int h = kvh * 4 + w;

    float4 qv = *(const float4*)(qbuf + (size_t)b * 4096 +
                                 (size_t)h * 128 + 4 * lane);
    qv.x *= INV_SQRT_D; qv.y *= INV_SQRT_D;
    qv.z *= INV_SQRT_D; qv.w *= INV_SQRT_D;

    const float* pbase = part + ((size_t)bk * 4 + w) * 8 * 132;
    float M = -INFINITY;
    #pragma unroll
    for (int s = 0; s < 8; ++s) M = fmaxf(M, pbase[s * 132 + 128]);
    float L = 0.0f, A0 = 0, A1 = 0, A2 = 0, A3 = 0;
    #pragma unroll
    for (int s = 0; s < 8; ++s) {
        const float cc = __expf(pbase[s * 132 + 128] - M);
        L  += pbase[s * 132 + 129] * cc;
        A0 += pbase[s * 132 + 4 * lane + 0] * cc;
        A1 += pbase[s * 132 + 4 * lane + 1] * cc;
        A2 += pbase[s * 132 + 4 * lane + 2] * cc;
        A3 += pbase[s * 132 + 4 * lane + 3] * cc;
    }

    // appended key at position 4096
    const size_t ooff = ((size_t)bk * 4097 + 4096) * 128 + 4 * lane;
    const float4 k4 = *(const float4*)(ko + ooff);
    const float4 v4 = *(const float4*)(vo + ooff);
    float s = qv.x * k4.x + qv.y * k4.y + qv.z * k4.z + qv.w * k4.w;
    s += __shfl_xor(s, 16); s += __shfl_xor(s, 8);
    s += __shfl_xor(s, 4);  s += __shfl_xor(s, 2);
    s += __shfl_xor(s, 1);
    const float mv = maskp[(size_t)b * 4097 + 4096];
    s += (mv == 1.0f) ? 0.0f : NEG_MIN_F;

    const float mn = fmaxf(M, s);
    const float cc = __expf(M - mn);
    const float p  = __expf(s - mn);
    L = L * cc + p;
    const float invL = 1.0f / L;
    float4 o;
    o.x = (A0 * cc + p * v4.x) * invL;
    o.y = (A1 * cc + p * v4.y) * invL;
    o.z = (A2 * cc + p * v4.z) * invL;
    o.w = (A3 * cc + p * v4.w) * invL;
    *(float4*)(attnbuf + (size_t)b * 4096 + (size_t)h * 128 + 4 * lane) = o;
}

// ---------------------------------------------------------------------------
extern "C" void kernel_launch(void* const* d_in, const int* in_sizes, int n_in,
                              void* d_out, int out_size, void* d_ws, size_t ws_size,
                              hipStream_t stream)
{
    (void)in_sizes; (void)n_in; (void)out_size; (void)ws_size;

    const float* x     = (const float*)d_in[1];
    const int*   pos   = (const int*)  d_in[2];
    const float* amask = (const float*)d_in[3];
    const float* cosT  = (const float*)d_in[4];
    const float* sinT  = (const float*)d_in[5];
    const float* kc    = (const float*)d_in[6];
    const float* vc    = (const float*)d_in[7];
    const float* Wq    = (const float*)d_in[8];
    const float* bq    = (const float*)d_in[9];
    const float* Wk    = (const float*)d_in[10];
    const float* bk    = (const float*)d_in[11];
    const float* Wv    = (const float*)d_in[12];
    const float* bv    = (const float*)d_in[13];
    const float* Wo    = (const float*)d_in[14];
    const float* bo    = (const float*)d_in[15];

    float* out  = (float*)d_out;                 // (8, 4096)
    float* kout = out + 32768;                   // (8,8,4097,128)
    float* vout = kout + (size_t)64 * 4097 * 128;

    float* ws      = (float*)d_ws;
    float* qbuf    = ws;              // 8*4096
    float* kbuf    = ws + 32768;      // 8*1024
    float* vbuf    = ws + 40960;      // 8*1024
    float* attnbuf = ws + 49152;      // 8*4096
    float* part    = ws + 81920;      // 64*4*8*132

    gemm8_wmma<<<256, 256, 0, stream>>>(x, Wq, bq, qbuf, 4096);
    gemm8_wmma<<<64,  256, 0, stream>>>(x, Wk, bk, kbuf, 1024);
    gemm8_wmma<<<64,  256, 0, stream>>>(x, Wv, bv, vbuf, 1024);
    rope_append<<<8, 256, 0, stream>>>(qbuf, kbuf, vbuf, pos, cosT, sinT, kout, vout);
    attn_partial<<<512, 256, 0, stream>>>(kc, vc, kout, vout, qbuf, amask, part);
    attn_reduce<<<64, 128, 0, stream>>>(part, qbuf, amask, kout, vout, attnbuf);
    gemm8_wmma<<<256, 256, 0, stream>>>(attnbuf, Wo, bo, out, 4096);
}